// HybridQuanvolution_65481071407379
// MI455X (gfx1250) — compile-verified
//
#include <hip/hip_runtime.h>
#include <math.h>

// out[b, j, 0] depends only on j%2  ->  entire output is {v0, v1} interleaved.
// Kernel 1 computes v0/v1 from the tiny parameter arrays (runs once, 1 wave).
// Kernel 2 is a pure store-bandwidth interleaved fill using the CDNA5 async
// LDS->global store engine (GLOBAL_STORE_ASYNC_FROM_LDS_B128, ASYNCcnt).

typedef float v4f __attribute__((ext_vector_type(4)));

__global__ void qf_params_kernel(const float* __restrict__ ry,      // [4]
                                 const float* __restrict__ rx,      // [8]
                                 const float* __restrict__ W,       // [3][2][2]
                                 const float* __restrict__ bb,      // [3][2]
                                 const float* __restrict__ sc,      // [3][2]
                                 const float* __restrict__ sh,      // [3][2]
                                 const float* __restrict__ outW,    // [2]
                                 const float* __restrict__ outB,    // [1]
                                 float* __restrict__ vals)          // [2] out
{
    if (threadIdx.x != 0 || blockIdx.x != 0) return;

    // Bloch algebra: <Z_0> = cos(ry0)*cos(sum rx), <Z_i> = cos(ry_i)
    float s = 0.0f;
#pragma unroll
    for (int i = 0; i < 8; ++i) s += rx[i];

    const float p0 = cosf(ry[0]) * cosf(s);
    const float p1 = cosf(ry[1]);
    const float p2 = cosf(ry[2]);
    const float p3 = cosf(ry[3]);

    // Two independent 2-vectors: (p0,p1) for even pairs, (p2,p3) for odd pairs.
    float h[2][2] = { { p0, p1 }, { p2, p3 } };

#pragma unroll
    for (int l = 0; l < 3; ++l) {
        const float w00 = W[l * 4 + 0], w01 = W[l * 4 + 1];
        const float w10 = W[l * 4 + 2], w11 = W[l * 4 + 3];
        const float b0 = bb[l * 2 + 0], b1 = bb[l * 2 + 1];
        const float s0 = sc[l * 2 + 0], s1 = sc[l * 2 + 1];
        const float f0 = sh[l * 2 + 0], f1 = sh[l * 2 + 1];
#pragma unroll
        for (int v = 0; v < 2; ++v) {
            const float a = h[v][0], b2 = h[v][1];
            const float t0 = tanhf(w00 * a + w01 * b2 + b0) * s0 + f0;
            const float t1 = tanhf(w10 * a + w11 * b2 + b1) * s1 + f1;
            h[v][0] = t0;
            h[v][1] = t1;
        }
    }

    const float ow0 = outW[0], ow1 = outW[1], ob = outB[0];
#pragma unroll
    for (int v = 0; v < 2; ++v) {
        const float logit = ow0 * h[v][0] + ow1 * h[v][1] + ob;
        vals[v] = logf(logit);
    }
}

// Bulk fill via the CDNA5 Tensor/async data path: the {v0,v1,v0,v1} pattern is
// staged in LDS once per block; every lane then issues
// GLOBAL_STORE_ASYNC_FROM_LDS_B128 — the async engine reads 16 B from LDS
// (all-lane broadcast, conflict-free) and writes them to the lane's global
// address.  Tracked by ASYNCcnt; completion enforced with s_wait_asynccnt.
__global__ __launch_bounds__(256) void fill_async_kernel(
    const float* __restrict__ vals,
    float* __restrict__ out,
    int n /* total f32 elements */)
{
    __shared__ v4f pat;

    const float v0 = vals[0];
    const float v1 = vals[1];
    if (threadIdx.x == 0) {
        pat = (v4f){ v0, v1, v0, v1 };
    }
    __syncthreads();

    // LDS byte offset of the pattern: AS3->generic casts on AMDGPU place the
    // LDS offset in the low 32 bits of the generic pointer.
    const unsigned int lds_off = (unsigned int)(unsigned long long)(&pat);

    const int n4     = n >> 2;                     // # of 16-byte slots
    const int tid    = blockIdx.x * blockDim.x + threadIdx.x;
    const int stride = gridDim.x * blockDim.x;

    // Launch sizing makes the grid cover n4 exactly 4x per thread with no
    // divergence, so EXEC is all-1s for every async store.
    for (int i = tid; i < n4; i += stride) {
        const unsigned long long gaddr =
            (unsigned long long)(out) + ((unsigned long long)i << 4);
        asm volatile("global_store_async_from_lds_b128 %0, %1, off"
                     :
                     : "v"(gaddr), "v"(lds_off)
                     : "memory");
    }

    // Drain the async engine before the wave retires (S_ENDPGM also implies
    // wait-idle, but be explicit about the ASYNCcnt dependency).
    asm volatile("s_wait_asynccnt 0x0" ::: "memory");

    // Scalar tail (out_size here is divisible by 4; kept generic) — plain
    // VGPR stores, tracked by STOREcnt.
    const int rem_base = n4 << 2;
    const int rem      = n - rem_base;
    if (tid < rem) {
        const int idx = rem_base + tid;
        out[idx] = (idx & 1) ? v1 : v0;
    }
}

extern "C" void kernel_launch(void* const* d_in, const int* in_sizes, int n_in,
                              void* d_out, int out_size, void* d_ws, size_t ws_size,
                              hipStream_t stream) {
    // setup_inputs() order: x, ry, rx_thetas, head_W, head_b, head_scale,
    //                       head_shift, out_W, out_b.  x (d_in[0]) is unused:
    // the quantum circuit ignores the patch data entirely.
    const float* ry  = (const float*)d_in[1];
    const float* rx  = (const float*)d_in[2];
    const float* W   = (const float*)d_in[3];
    const float* bb  = (const float*)d_in[4];
    const float* sc  = (const float*)d_in[5];
    const float* sh  = (const float*)d_in[6];
    const float* oW  = (const float*)d_in[7];
    const float* oB  = (const float*)d_in[8];

    float* vals = (float*)d_ws;   // 2 floats of scratch

    qf_params_kernel<<<1, 32, 0, stream>>>(ry, rx, W, bb, sc, sh, oW, oB, vals);

    const int n  = out_size;                 // 12,845,056 f32 = 51.4 MB
    const int n4 = n >> 2;
    // 3136 blocks x 256 threads x 4 async b128 stores/thread covers n4
    // exactly: ~25K wave32 waves — enough to saturate 23.3 TB/s of store
    // bandwidth, few enough that wave-launch overhead is negligible.
    int blocks = (n4 + 1023) / 1024;
    if (blocks < 1) blocks = 1;
    fill_async_kernel<<<blocks, 256, 0, stream>>>(vals, (float*)d_out, n);
}